// ConvModule_35905926594660
// MI455X (gfx1250) — compile-verified
//
#include <hip/hip_runtime.h>
#include <hip/hip_bf16.h>

// ---------------------------------------------------------------------------
// Bidirectional SAGEConv (mean aggr) for MI455X / gfx1250.
//   out = mean_in(x) @ Wl_in^T + mean_out(x) @ Wl_out^T
//       + x @ (Wr_in + Wr_out)^T + (b_in + b_out)
// Phase 1: fp32 atomic scatter of edge messages into L2-resident accumulators
//          (x is 51.2 MB -> fully L2-resident on the 192 MB L2).
// Phase 2: fp32 WMMA GEMM, K = 3 x 128, using V_WMMA_F32_16X16X4_F32 so the
//          result keeps full fp32 precision (GEMM is only ~10 GFLOP; the
//          scatter atomics dominate, so there is nothing to gain from fp16).
// ---------------------------------------------------------------------------

#define N_NODES 100000
#define N_EDGES 640000
#define D 128

typedef float v2f __attribute__((ext_vector_type(2)));
typedef float v8f __attribute__((ext_vector_type(8)));

__device__ __forceinline__ void atomAddF(float* p, float v) {
    (void)__hip_atomic_fetch_add(p, v, __ATOMIC_RELAXED, __HIP_MEMORY_SCOPE_AGENT);
}

// ---------------------------------------------------------------------------
// Kernel 0: zero the aggregation scratch; fold Wr_in+Wr_out and the biases.
// ---------------------------------------------------------------------------
__global__ __launch_bounds__(256) void sage_init(
    float* __restrict__ agg_in, float* __restrict__ agg_out,
    float* __restrict__ cnt_in, float* __restrict__ cnt_out,
    const float* __restrict__ w_r_in, const float* __restrict__ w_r_out,
    const float* __restrict__ b_l_in, const float* __restrict__ b_l_out,
    float* __restrict__ w_sum, float* __restrict__ b_sum)
{
    const int tid    = blockIdx.x * blockDim.x + threadIdx.x;
    const int stride = gridDim.x * blockDim.x;
    const int nAgg = N_NODES * D;
    for (int j = tid; j < nAgg; j += stride) { agg_in[j] = 0.0f; agg_out[j] = 0.0f; }
    for (int j = tid; j < N_NODES; j += stride) { cnt_in[j] = 0.0f; cnt_out[j] = 0.0f; }
    for (int j = tid; j < D * D; j += stride) w_sum[j] = w_r_in[j] + w_r_out[j];
    for (int j = tid; j < D; j += stride)     b_sum[j] = b_l_in[j] + b_l_out[j];
}

// ---------------------------------------------------------------------------
// Kernel 1: edge scatter. One wave32 per edge; each lane owns 4 features.
//   agg_in[dst]  += x[src]   (forward conv)
//   agg_out[src] += x[dst]   (reverse conv)
// ---------------------------------------------------------------------------
__global__ __launch_bounds__(256) void sage_scatter(
    const int* __restrict__ src, const int* __restrict__ dst,
    const float* __restrict__ x,
    float* __restrict__ agg_in, float* __restrict__ agg_out,
    float* __restrict__ cnt_in, float* __restrict__ cnt_out)
{
    const int e = blockIdx.x * 8 + (threadIdx.x >> 5);   // 8 waves / block
    if (e >= N_EDGES) return;
    const int lane = threadIdx.x & 31;
    const int s = src[e];
    const int d = dst[e];

    const float4 xs = *(const float4*)(x + (size_t)s * D + lane * 4);
    const float4 xd = *(const float4*)(x + (size_t)d * D + lane * 4);

    float* pin  = agg_in  + (size_t)d * D + lane * 4;
    float* pout = agg_out + (size_t)s * D + lane * 4;
    atomAddF(pin + 0, xs.x); atomAddF(pin + 1, xs.y);
    atomAddF(pin + 2, xs.z); atomAddF(pin + 3, xs.w);
    atomAddF(pout + 0, xd.x); atomAddF(pout + 1, xd.y);
    atomAddF(pout + 2, xd.z); atomAddF(pout + 3, xd.w);

    if (lane == 0) { atomAddF(cnt_in + d, 1.0f); atomAddF(cnt_out + s, 1.0f); }
}

// ---------------------------------------------------------------------------
// Kernel 2: WMMA GEMM.
// V_WMMA_F32_16X16X4_F32 fragments (wave32):
//   A 16x4 : lanes 0-15 -> M=lane,    K={k,k+1};  lanes 16-31 -> M=lane-16, K={k+2,k+3}
//   B 4x16 : lanes 0-15 -> N=lane,    K={k,k+1};  lanes 16-31 -> N=lane-16, K={k+2,k+3}
//   C 16x16: acc[r] -> lanes 0-15: (M=r, N=lane); lanes 16-31: (M=8+r, N=lane-16)
// B[k][n] = W[n][k], so each lane's B fragment is a contiguous float2 of row W[col].
// ---------------------------------------------------------------------------
__device__ __forceinline__ v8f mac_section(const float* __restrict__ A,
                                           const float* __restrict__ W,
                                           float sc, v8f acc)
{
#pragma unroll
    for (int k = 0; k < D; k += 4) {
        v2f a = *(const v2f*)(A + k);
        a.x *= sc; a.y *= sc;
        v2f b = *(const v2f*)(W + k);
        acc = __builtin_amdgcn_wmma_f32_16x16x4_f32(
            /*neg_a=*/false, a, /*neg_b=*/false, b,
            /*c_mod=*/(short)0, acc, /*reuse_a=*/false, /*reuse_b=*/false);
    }
    return acc;
}

__global__ __launch_bounds__(256) void sage_gemm(
    const float* __restrict__ x,
    const float* __restrict__ agg_in, const float* __restrict__ agg_out,
    const float* __restrict__ cnt_in, const float* __restrict__ cnt_out,
    const float* __restrict__ w_l_in, const float* __restrict__ w_l_out,
    const float* __restrict__ w_sum,  const float* __restrict__ b_sum,
    float* __restrict__ out)
{
    const int mtile = blockIdx.x;            // 16 output rows per block
    const int wave  = threadIdx.x >> 5;      // 8 waves -> 8 N-tiles of 16
    const int lane  = threadIdx.x & 31;
    const int half  = lane >> 4;
    const int l15   = lane & 15;
    const int row   = mtile * 16 + l15;      // M row held by this lane (A frag)
    const int col   = wave * 16 + l15;       // N col held by this lane (B/C frag)

    const float sc_in  = 1.0f / fmaxf(cnt_in[row],  1.0f);  // mean; isolated -> /1
    const float sc_out = 1.0f / fmaxf(cnt_out[row], 1.0f);

    const float bias = b_sum[col];
    v8f acc;
#pragma unroll
    for (int r = 0; r < 8; ++r) acc[r] = bias;

    const size_t aoff = (size_t)row * D + 2 * half;  // K offset per lane-half
    const size_t woff = (size_t)col * D + 2 * half;

    acc = mac_section(agg_in  + aoff, w_l_in  + woff, sc_in,  acc);
    acc = mac_section(agg_out + aoff, w_l_out + woff, sc_out, acc);
    acc = mac_section(x       + aoff, w_sum   + woff, 1.0f,   acc);

#pragma unroll
    for (int r = 0; r < 8; ++r) {
        const int m = mtile * 16 + half * 8 + r;
        out[(size_t)m * D + col] = acc[r];
    }
}

// ---------------------------------------------------------------------------
extern "C" void kernel_launch(void* const* d_in, const int* in_sizes, int n_in,
                              void* d_out, int out_size, void* d_ws, size_t ws_size,
                              hipStream_t stream)
{
    const float* x       = (const float*)d_in[0];
    const int*   ei      = (const int*)d_in[1];     // [2, E] row-major (int32)
    const float* w_l_in  = (const float*)d_in[2];
    const float* b_l_in  = (const float*)d_in[3];
    const float* w_r_in  = (const float*)d_in[4];
    const float* w_l_out = (const float*)d_in[5];
    const float* b_l_out = (const float*)d_in[6];
    const float* w_r_out = (const float*)d_in[7];
    float* out = (float*)d_out;

    // Workspace layout (all offsets 128B+ aligned):
    //   agg_in  [N,128] f32 | agg_out [N,128] f32 | cnt_in [N] | cnt_out [N]
    //   w_sum [128,128] | b_sum [128]
    const size_t NB   = (size_t)N_NODES * D * sizeof(float);   // 51,200,000
    const size_t CNTB = (size_t)N_NODES * sizeof(float);       //    400,000
    char* ws = (char*)d_ws;
    float* agg_in  = (float*)(ws);
    float* agg_out = (float*)(ws + NB);
    float* cnt_in  = (float*)(ws + 2 * NB);
    float* cnt_out = (float*)(ws + 2 * NB + CNTB);
    float* w_sum   = (float*)(ws + 2 * NB + 2 * CNTB);
    float* b_sum   = (float*)(ws + 2 * NB + 2 * CNTB + (size_t)D * D * sizeof(float));

    (void)in_sizes; (void)n_in; (void)out_size; (void)ws_size;

    sage_init<<<2048, 256, 0, stream>>>(agg_in, agg_out, cnt_in, cnt_out,
                                        w_r_in, w_r_out, b_l_in, b_l_out,
                                        w_sum, b_sum);

    sage_scatter<<<N_EDGES / 8, 256, 0, stream>>>(ei, ei + N_EDGES, x,
                                                  agg_in, agg_out, cnt_in, cnt_out);

    sage_gemm<<<N_NODES / 16, 256, 0, stream>>>(x, agg_in, agg_out, cnt_in, cnt_out,
                                                w_l_in, w_l_out, w_sum, b_sum, out);
}